// OutMultiChoices_22986664969012
// MI455X (gfx1250) — compile-verified
//
#include <hip/hip_runtime.h>
#include <math.h>

typedef float v2f __attribute__((ext_vector_type(2)));
typedef float v8f __attribute__((ext_vector_type(8)));

#define BATCH 8192
#define DIM   512
#define K4    2048          // 4*DIM
#define LQMAX 32
#define LAMAX 16
#define TT    32
#define BN_EPS 1e-5f

// ---------------------------------------------------------------------------
// Kernel 1: fused masked-mean / mean pooling of the four [L,B,D] tensors,
// writing directly into the concatenated [B, 4D] activation matrix.
// One thread per (b, d). Lanes within a wave share b (consecutive d) -> fully
// coalesced 128B requests per l-step; non-temporal loads keep the 1.88 GB
// stream out of L2 so the concat/h matrices stay resident there.
// ---------------------------------------------------------------------------
__global__ __launch_bounds__(256) void pool_concat_kernel(
    const float* __restrict__ qv,    // [LQ, B, D]
    const float* __restrict__ vqu,   // [T,  B, D]
    const int*   __restrict__ qlen,  // [B]
    const float* __restrict__ av,    // [LA, B, D]
    const float* __restrict__ van,   // [T,  B, D]
    const int*   __restrict__ alen,  // [B]
    float*       __restrict__ outc)  // [B, 4D]
{
    const int idx = blockIdx.x * 256 + threadIdx.x;      // 0 .. B*D-1
    const int b = idx >> 9;                              // D == 512
    const int d = idx & (DIM - 1);
    const long stride = (long)BATCH * DIM;
    const long base   = (long)b * DIM + d;

    const int ql = qlen[b];
    float s = 0.f;
    for (int l = 0; l < ql; ++l)
        s += __builtin_nontemporal_load(qv + base + (long)l * stride);
    const float q_m = s / (float)ql;

    s = 0.f;
#pragma unroll
    for (int l = 0; l < TT; ++l)
        s += __builtin_nontemporal_load(vqu + base + (long)l * stride);
    const float vq_m = s * (1.0f / TT);

    const int al = alen[b];
    s = 0.f;
    for (int l = 0; l < al; ++l)
        s += __builtin_nontemporal_load(av + base + (long)l * stride);
    const float a_m = s / (float)al;

    s = 0.f;
#pragma unroll
    for (int l = 0; l < TT; ++l)
        s += __builtin_nontemporal_load(van + base + (long)l * stride);
    const float va_m = s * (1.0f / TT);

    const long ob = (long)b * (4 * DIM) + d;
    outc[ob + 0 * DIM] = q_m;
    outc[ob + 1 * DIM] = vq_m;
    outc[ob + 2 * DIM] = a_m;
    outc[ob + 3 * DIM] = va_m;
}

// ---------------------------------------------------------------------------
// Kernel 2: h = gelu(out @ W1^T + b1) via V_WMMA_F32_16X16X4_F32.
// One wave owns a 16x32 tile of h (A fragment reused across 2 B tiles).
// A fragment (16x4 f32): lane(lo,hi) -> float2 { A[lo][k+2*hi], A[lo][k+2*hi+1] }
// B fragment (4x16 f32): lane(lo,hi) -> float2 { W1[n][k+2*hi], W1[n][k+2*hi+1] }
// C/D (16x16 f32): c[r] on lane(lo,hi) -> h[rowBase + r + 8*hi][colBase + lo]
// Also emits deterministic per-tile BatchNorm partials (sum, sum^2 per col).
// ---------------------------------------------------------------------------
__device__ __forceinline__ float gelu_exact(float x) {
    return 0.5f * x * (1.0f + erff(x * 0.70710678118654752440f));
}

__global__ __launch_bounds__(256) void gemm_gelu_kernel(
    const float* __restrict__ A,      // [B, 2048]
    const float* __restrict__ W1,     // [512, 2048] row-major
    const float* __restrict__ bias1,  // [512]
    float*       __restrict__ H,      // [B, 512]
    float*       __restrict__ psum,   // [512 rowtiles, 512]
    float*       __restrict__ psumsq) // [512 rowtiles, 512]
{
    const int wave = blockIdx.x * (256 / 32) + (threadIdx.x >> 5);
    const int lane = threadIdx.x & 31;
    const int lo = lane & 15;
    const int hi = lane >> 4;

    const int rt = wave & 511;        // 512 row tiles of 16
    const int cp = wave >> 9;         // 16 column pairs of 32
    const int rowBase = rt * 16;
    const int colBase = cp * 32;

    const float* arow  = A  + (long)(rowBase + lo)      * K4 + 2 * hi;
    const float* b0row = W1 + (long)(colBase + lo)      * K4 + 2 * hi;
    const float* b1row = W1 + (long)(colBase + 16 + lo) * K4 + 2 * hi;

    v8f c0 = {};
    v8f c1 = {};
#pragma unroll 4
    for (int k = 0; k < K4; k += 4) {
        v2f a   = *(const v2f*)(arow  + k);
        v2f bb0 = *(const v2f*)(b0row + k);
        v2f bb1 = *(const v2f*)(b1row + k);
        c0 = __builtin_amdgcn_wmma_f32_16x16x4_f32(false, a, false, bb0,
                                                   (short)0, c0, false, false);
        c1 = __builtin_amdgcn_wmma_f32_16x16x4_f32(false, a, false, bb1,
                                                   (short)0, c1, false, false);
    }

    const int col0 = colBase + lo;
    const int col1 = colBase + 16 + lo;
    const float bias0 = bias1[col0];
    const float biasb = bias1[col1];

    float s0 = 0.f, q0 = 0.f, s1 = 0.f, q1 = 0.f;
#pragma unroll
    for (int r = 0; r < 8; ++r) {
        const int row = rowBase + r + 8 * hi;
        const float g0 = gelu_exact(c0[r] + bias0);
        const float g1 = gelu_exact(c1[r] + biasb);
        H[(long)row * DIM + col0] = g0;
        H[(long)row * DIM + col1] = g1;
        s0 += g0; q0 += g0 * g0;
        s1 += g1; q1 += g1 * g1;
    }
    // fold the two 8-row halves of each column (lane L <-> L^16 share a column)
    s0 += __shfl_xor(s0, 16, 32);
    q0 += __shfl_xor(q0, 16, 32);
    s1 += __shfl_xor(s1, 16, 32);
    q1 += __shfl_xor(q1, 16, 32);
    // lanes 0-15 publish c0's columns, lanes 16-31 publish c1's columns
    const float sOut = hi ? s1 : s0;
    const float qOut = hi ? q1 : q0;
    psum  [(long)rt * DIM + colBase + lane] = sOut;
    psumsq[(long)rt * DIM + colBase + lane] = qOut;
}

// ---------------------------------------------------------------------------
// Kernel 3: deterministic fold of 512 row-tile partials per column ->
// BN scale/shift:  y = h*scale + shift,  scale = gamma*rsqrt(var+eps),
//                  shift = beta - mu*scale
// ---------------------------------------------------------------------------
__global__ __launch_bounds__(256) void bn_finalize_kernel(
    const float* __restrict__ psum,
    const float* __restrict__ psumsq,
    const float* __restrict__ gamma,
    const float* __restrict__ beta,
    float*       __restrict__ scale,
    float*       __restrict__ shift)
{
    const int col = blockIdx.x * 256 + threadIdx.x;   // grid 2 -> 512 cols
    float s = 0.f, q = 0.f;
    for (int rtile = 0; rtile < 512; ++rtile) {
        s += psum  [(long)rtile * DIM + col];
        q += psumsq[(long)rtile * DIM + col];
    }
    const float mu  = s * (1.0f / BATCH);
    const float var = q * (1.0f / BATCH) - mu * mu;
    const float inv = rsqrtf(var + BN_EPS);
    const float sc  = gamma[col] * inv;
    scale[col] = sc;
    shift[col] = beta[col] - mu * sc;
}

// ---------------------------------------------------------------------------
// Kernel 4: logits[b] = sum_n (h[b][n]*scale[n]+shift[n]) * W2[n] + b2
// One wave32 per row, shuffle tree reduction.
// ---------------------------------------------------------------------------
__global__ __launch_bounds__(256) void bn_logits_kernel(
    const float* __restrict__ H,
    const float* __restrict__ scale,
    const float* __restrict__ shift,
    const float* __restrict__ W2,    // [1, 512]
    const float* __restrict__ b2,    // [1]
    float*       __restrict__ out)   // [B]
{
    const int row  = blockIdx.x * 8 + (threadIdx.x >> 5);
    const int lane = threadIdx.x & 31;
    const float* hrow = H + (long)row * DIM;

    float acc = 0.f;
#pragma unroll
    for (int i = 0; i < DIM / 32; ++i) {
        const int n = i * 32 + lane;
        const float v = hrow[n] * scale[n] + shift[n];
        acc += v * W2[n];
    }
#pragma unroll
    for (int off = 16; off > 0; off >>= 1)
        acc += __shfl_xor(acc, off, 32);
    if (lane == 0)
        out[row] = acc + b2[0];
}

// ---------------------------------------------------------------------------
extern "C" void kernel_launch(void* const* d_in, const int* in_sizes, int n_in,
                              void* d_out, int out_size, void* d_ws, size_t ws_size,
                              hipStream_t stream) {
    const float* qv    = (const float*)d_in[0];
    const float* vqu   = (const float*)d_in[1];
    const int*   qlen  = (const int*)  d_in[2];
    const float* av    = (const float*)d_in[3];
    const float* van   = (const float*)d_in[4];
    const int*   alen  = (const int*)  d_in[5];
    const float* W1    = (const float*)d_in[6];
    const float* b1    = (const float*)d_in[7];
    const float* gamma = (const float*)d_in[8];
    const float* beta  = (const float*)d_in[9];
    const float* W2    = (const float*)d_in[10];
    const float* b2    = (const float*)d_in[11];

    char* ws = (char*)d_ws;
    float* outc   = (float*)ws;                                        // 64 MB
    float* H      = (float*)(ws + (size_t)BATCH * 4 * DIM * 4);        // +16 MB
    float* psum   = (float*)(ws + (size_t)BATCH * 5 * DIM * 4);        // +1 MB
    float* psumsq = psum   + 512 * 512;                                // +1 MB
    float* scale  = psumsq + 512 * 512;
    float* shift  = scale  + 512;

    pool_concat_kernel<<<(BATCH * DIM) / 256, 256, 0, stream>>>(
        qv, vqu, qlen, av, van, alen, outc);

    // 8192 waves (512 row tiles x 16 col pairs), 8 waves per block
    gemm_gelu_kernel<<<8192 / 8, 256, 0, stream>>>(
        outc, W1, b1, H, psum, psumsq);

    bn_finalize_kernel<<<2, 256, 0, stream>>>(
        psum, psumsq, gamma, beta, scale, shift);

    bn_logits_kernel<<<BATCH / 8, 256, 0, stream>>>(
        H, scale, shift, W2, b2, (float*)d_out);
}